// TemporalConv_15882789060657
// MI455X (gfx1250) — compile-verified
//
#include <hip/hip_runtime.h>
#include <hip/hip_bf16.h>
#include <math.h>

typedef __attribute__((ext_vector_type(2))) float v2f;
typedef __attribute__((ext_vector_type(8))) float v8f;

#define N_NODES 10000
#define N_EDGES 80000
#define B_DIM   4
#define T_DIM   12
#define M_DIM   48              // B*T
#define C_IN    64
#define C_OUT   128
#define ZELEMS  (M_DIM * C_IN)  // 3072 floats per node
#define ZTOTAL  ((size_t)N_NODES * ZELEMS)
#define NODES_PER_BLOCK 8

// Packed-W LDS layout: pair row p = km*32 + k/2 holds float2{W[k][c],W[k+1][c]}.
// PAIR_STRIDE=144 float2 (=288 dwords, ==32 mod 64) so the two lane-halves of a
// wave (which read adjacent pair rows) hit disjoint LDS bank halves.
#define N_PAIRS     (3 * 32)    // 3 Chebyshev mats * 32 K-pairs
#define PAIR_STRIDE 144
#define SW_FLOATS   (N_PAIRS * PAIR_STRIDE * 2)   // 27648 floats = 110.6 KB

// --- 1. x(B,C,N,T) -> Z0(N, M=b*T+t, C); also zero Z1 accumulator ---
__global__ void prep_kernel(const float* __restrict__ x,
                            float* __restrict__ Z0,
                            float* __restrict__ Z1) {
    size_t i = (size_t)blockIdx.x * blockDim.x + threadIdx.x;
    if (i >= ZTOTAL) return;
    int c = (int)(i & (C_IN - 1));
    int m = (int)((i >> 6) % M_DIM);
    int n = (int)(i / ZELEMS);
    int b = m / T_DIM;
    int t = m - b * T_DIM;
    Z0[i] = x[(((size_t)(b * C_IN + c)) * N_NODES + n) * T_DIM + t];
    Z1[i] = 0.0f;
}

// --- 2. Z2 = -Z0 (so Z2 += 2*w*Z1[src] yields 2*spmm(Z1) - Z0) ---
__global__ void negcopy_kernel(const float* __restrict__ Z0,
                               float* __restrict__ Z2) {
    size_t i = (size_t)blockIdx.x * blockDim.x + threadIdx.x;
    if (i >= ZTOTAL) return;
    Z2[i] = -Z0[i];
}

// --- 3. SpMM scatter-add: one edge per block, hardware f32 atomics ---
__global__ void spmm_kernel(const float* __restrict__ Zs,
                            float* __restrict__ Zd,
                            const int* __restrict__ esrc,
                            const int* __restrict__ edst,
                            const float* __restrict__ ew,
                            float scale) {
    int e = blockIdx.x;
    float w = ew[e] * scale;
    const float* __restrict__ ps = Zs + (size_t)esrc[e] * ZELEMS;
    float*       __restrict__ pd = Zd + (size_t)edst[e] * ZELEMS;
    for (int i = threadIdx.x; i < ZELEMS; i += blockDim.x) {
        unsafeAtomicAdd(pd + i, w * ps[i]);   // global_atomic_add_f32 (no return)
    }
}

// --- 4. Per-node GEMM via V_WMMA_F32_16X16X4_F32 + gated epilogue ---
// 256 threads = 8 wave32; wave w owns output-channel tile [16w,16w+16).
// W staged once per block into LDS (paired for single ds_load_b64 per WMMA).
// acc(48x128) staged in LDS so p (c) and q (c+64) channels can be paired.
__global__ void __launch_bounds__(256)
gemm_epilogue_kernel(const float* __restrict__ Z0,
                     const float* __restrict__ Z1,
                     const float* __restrict__ Z2,
                     const float* __restrict__ W,
                     const float* __restrict__ bias,
                     const float* __restrict__ x,
                     float* __restrict__ out) {
    __shared__ __align__(16) float s_w[SW_FLOATS];   // 110.6 KB packed W
    __shared__ float s_acc[M_DIM * C_OUT];           // 6144 floats = 24 KB
    __shared__ float s_bias[C_OUT];

    const int tid  = threadIdx.x;
    const int wv   = tid >> 5;         // wave id 0..7 -> col tile
    const int lane = tid & 31;
    const int row  = lane & 15;        // A/B: M or N index within tile
    const int kh   = (lane >> 4) << 1; // K sub-offset 0 or 2 (lane halves)
    const int n0   = wv << 4;          // output channel tile base

    // Cooperative stage of W (3*64*128 floats) into paired LDS layout.
    for (int idx = tid; idx < 3 * C_IN * C_OUT; idx += 256) {
        const int km = idx >> 13;           // / 8192
        const int rem = idx & 8191;
        const int k = rem >> 7;             // / 128
        const int c = rem & 127;
        const int p = km * 32 + (k >> 1);
        s_w[(p * PAIR_STRIDE + c) * 2 + (k & 1)] = W[idx];
    }
    if (tid < C_OUT) s_bias[tid] = bias[tid];
    __syncthreads();

    const float* const Zs[3] = {Z0, Z1, Z2};

    for (int it = 0; it < NODES_PER_BLOCK; ++it) {
        const int n = blockIdx.x * NODES_PER_BLOCK + it;   // block-uniform
        if (n >= N_NODES) break;

        #pragma unroll
        for (int rt = 0; rt < 3; ++rt) {                   // row tiles of M=48
            v8f acc = {};
            #pragma unroll
            for (int km = 0; km < 3; ++km) {               // Chebyshev terms
                const float* __restrict__ Zn =
                    Zs[km] + (size_t)n * ZELEMS + (size_t)(rt * 16) * C_IN;
                #pragma unroll
                for (int ks = 0; ks < 16; ++ks) {          // K=64 in steps of 4
                    const int kb = (ks << 2) + kh;
                    // A(16x4): lane<16 -> K=kb,kb+1 ; lanes 16-31 -> kb+2 (via kh)
                    v2f a = *reinterpret_cast<const v2f*>(&Zn[row * C_IN + kb]);
                    // B(4x16): one ds_load_b64 gives K-rows kb,kb+1 at col n0+row
                    const int p = km * 32 + (kb >> 1);
                    v2f bb = *reinterpret_cast<const v2f*>(
                        &s_w[(p * PAIR_STRIDE + n0 + row) * 2]);
                    acc = __builtin_amdgcn_wmma_f32_16x16x4_f32(
                        false, a, false, bb, (short)0, acc, false, false);
                }
            }
            // D layout: VGPR v, lane L -> M = v + (L>=16 ? 8:0), N = L&15
            const int mbase = rt * 16 + ((lane >> 4) << 3);
            const float bval = s_bias[n0 + row];
            #pragma unroll
            for (int v = 0; v < 8; ++v)
                s_acc[(mbase + v) * C_OUT + n0 + row] = acc[v] + bval;
        }
        __syncthreads();

        // epilogue: out[b,c,n,t] = (p + x) * sigmoid(q), 3072 outputs
        for (int i = tid; i < M_DIM * C_IN; i += 256) {
            const int c = i / M_DIM;            // m fastest -> t contiguous stores
            const int m = i - c * M_DIM;
            const int b = m / T_DIM;
            const int t = m - b * T_DIM;
            const float p = s_acc[m * C_OUT + c];
            const float q = s_acc[m * C_OUT + c + C_IN];
            const size_t xi = (((size_t)(b * C_IN + c)) * N_NODES + n) * T_DIM + t;
            const float xv = x[xi];
            out[xi] = (p + xv) * (1.0f / (1.0f + __expf(-q)));
        }
        __syncthreads();
    }
}

extern "C" void kernel_launch(void* const* d_in, const int* in_sizes, int n_in,
                              void* d_out, int out_size, void* d_ws, size_t ws_size,
                              hipStream_t stream) {
    const float* x    = (const float*)d_in[0];
    const float* ew   = (const float*)d_in[1];
    const float* W    = (const float*)d_in[2];
    const float* bias = (const float*)d_in[3];
    const int*   esrc = (const int*)d_in[4];   // int32 per harness convention
    const int*   edst = (const int*)d_in[5];
    float* out = (float*)d_out;

    float* Z0 = (float*)d_ws;
    float* Z1 = Z0 + ZTOTAL;
    float* Z2 = Z1 + ZTOTAL;

    const int eltBlocks = (int)((ZTOTAL + 255) / 256);

    // Z0 = xn, Z1 = 0
    prep_kernel<<<eltBlocks, 256, 0, stream>>>(x, Z0, Z1);
    // Z1 = spmm(Z0)
    spmm_kernel<<<N_EDGES, 256, 0, stream>>>(Z0, Z1, esrc, edst, ew, 1.0f);
    // Z2 = -Z0, then Z2 += 2*spmm(Z1)
    negcopy_kernel<<<eltBlocks, 256, 0, stream>>>(Z0, Z2);
    spmm_kernel<<<N_EDGES, 256, 0, stream>>>(Z1, Z2, esrc, edst, ew, 2.0f);
    // acc = sum_k Zk@Wk + b ; out = (p + x) * sigmoid(q)
    gemm_epilogue_kernel<<<N_NODES / NODES_PER_BLOCK, 256, 0, stream>>>(
        Z0, Z1, Z2, W, bias, x, out);
}